// SRCTModel_5652176962056
// MI455X (gfx1250) — compile-verified
//
#include <hip/hip_runtime.h>
#include <math.h>

// ---- problem constants (from reference) ----
#define S_CNT   10000
#define R_CNT   10000
#define K_SR    64      // K_S == K_R == 64
#define K_PDIM  128
#define ROWPAD  132     // 128 + 4 pad floats -> conflict-free fragment reads

typedef float v2f __attribute__((ext_vector_type(2)));
typedef float v8f __attribute__((ext_vector_type(8)));

#define WAVES_PER_BLOCK 2
#define THREADS_PER_BLOCK (WAVES_PER_BLOCK * 32)
#define SAMPLES_PER_WAVE 16
#define SAMPLES_PER_BLOCK (WAVES_PER_BLOCK * SAMPLES_PER_WAVE)

__global__ __launch_bounds__(THREADS_PER_BLOCK)
void srct_wmma_kernel(const int*   __restrict__ X,         // (N,4) int32: s,r,p,t
                      const float* __restrict__ s_embeds,  // (T*S_CNT, 64)
                      const float* __restrict__ r_embeds,  // (T*R_CNT, 64)
                      const float* __restrict__ p_embeds,  // (P_CNT, 128)
                      float*       __restrict__ out,       // (N,)
                      int n)
{
    // Per-wave tiles: A[16][128] = concat(se,re) rows, B[16][128] = pe rows.
    __shared__ float ldsA[WAVES_PER_BLOCK][SAMPLES_PER_WAVE][ROWPAD];
    __shared__ float ldsB[WAVES_PER_BLOCK][SAMPLES_PER_WAVE][ROWPAD];

    const int tid  = (int)threadIdx.x;
    const int wave = tid >> 5;          // gfx1250 is wave32
    const int lane = tid & 31;

    const int tileBase = ((int)blockIdx.x * WAVES_PER_BLOCK + wave) * SAMPLES_PER_WAVE;

    // ---------------- Phase 1: gather rows into LDS ----------------
    // lane -> (sample j = lane/2, half h = lane&1). h==0: se / pe[0:64],
    // h==1: re / pe[64:128]. 16 x global_load_b128 per lane per matrix.
    const int j = lane >> 1;
    const int h = lane & 1;
    int sidx = tileBase + j;
    if (sidx >= n) sidx = n - 1;        // clamp (stores are guarded; keeps EXEC full)

    const int4 xr = *(const int4*)(X + 4LL * sidx);
    const int s = xr.x, r = xr.y, p = xr.z, t = xr.w;

    const float* arow = h ? (r_embeds + (long long)(r + t * R_CNT) * K_SR)
                          : (s_embeds + (long long)(s + t * S_CNT) * K_SR);
    const float* brow = p_embeds + (long long)p * K_PDIM + h * K_SR;

    float* la = &ldsA[wave][j][h * K_SR];
    float* lb = &ldsB[wave][j][h * K_SR];
#pragma unroll
    for (int q = 0; q < K_SR / 4; ++q) {
        *(float4*)(la + 4 * q) = *(const float4*)(arow + 4 * q);
        *(float4*)(lb + 4 * q) = *(const float4*)(brow + 4 * q);
    }

    // Each wave reads only LDS it wrote itself; DS ops are in-order per wave.
    // Drain the DS queue (inline asm: no builtin for s_wait_dscnt) and fence
    // the compiler so fragment reads can't be scheduled above the stores.
    asm volatile("s_wait_dscnt 0x0" ::: "memory");
    __builtin_amdgcn_wave_barrier();

    // ---------------- Phase 2: WMMA K-loop (f32 16x16x4) ----------------
    // A-frag (16x4 f32, ISA 7.12.2): lanes 0-15 row M=lane hold {K+0,K+1},
    // lanes 16-31 row M=lane-16 hold {K+2,K+3}. B-frag (4x16) is the mirror:
    // VGPR0 = {K+0 | K+2}, VGPR1 = {K+1 | K+3} striped over N=0..15 per half.
    // Both reduce to a float2 ds_load at lds[lane&15][k + (lane<16 ? 0 : 2)].
    const int row  = lane & 15;
    const int koff = (lane < 16) ? 0 : 2;
    const float* aPtr = &ldsA[wave][row][koff];
    const float* bPtr = &ldsB[wave][row][koff];

    v8f c = {0.f, 0.f, 0.f, 0.f, 0.f, 0.f, 0.f, 0.f};
#pragma unroll
    for (int k = 0; k < K_PDIM; k += 4) {
        v2f a = *(const v2f*)(aPtr + k);
        v2f b = *(const v2f*)(bPtr + k);
        // 8 args: (neg_a, A, neg_b, B, c_mod, C, reuse_a, reuse_b)
        c = __builtin_amdgcn_wmma_f32_16x16x4_f32(
                false, a, false, b, (short)0, c, false, false);
    }

    // ---------------- Phase 3: diagonal extract + sigmoid ----------------
    // C/D layout: lanes 0-15 elem e -> D[e][lane]; lanes 16-31 elem e -> D[e+8][lane-16].
    // D[i][i]: i<8 -> lane i, elem i;  i>=8 -> lane i+16 (24..31), elem i-8.
    const bool active = (lane < 8) || (lane >= 24);
    const int  e      = (lane < 8) ? lane : (lane - 24);   // 0..7 when active
    float v = c[0];
#pragma unroll
    for (int q = 1; q < 8; ++q)
        if (e == q) v = c[q];

    const int sample = (lane < 8) ? lane : (lane - 16);
    const int oidx   = tileBase + sample;
    if (active && oidx < n) {
        out[oidx] = 1.0f / (1.0f + expf(-v));
    }
}

extern "C" void kernel_launch(void* const* d_in, const int* in_sizes, int n_in,
                              void* d_out, int out_size, void* d_ws, size_t ws_size,
                              hipStream_t stream)
{
    const int*   X        = (const int*)  d_in[0];   // (N,4)
    const float* s_embeds = (const float*)d_in[1];
    const float* r_embeds = (const float*)d_in[2];
    const float* p_embeds = (const float*)d_in[3];
    float*       out      = (float*)d_out;

    const int n    = in_sizes[0] / 4;                // N samples
    const int grid = (n + SAMPLES_PER_BLOCK - 1) / SAMPLES_PER_BLOCK;

    srct_wmma_kernel<<<grid, THREADS_PER_BLOCK, 0, stream>>>(
        X, s_embeds, r_embeds, p_embeds, out, n);
}